// IterativeModule_33397665694045
// MI455X (gfx1250) — compile-verified
//
#include <hip/hip_runtime.h>
#include <cstdint>

#define BLOCK 256
#define WAVES_PER_BLOCK (BLOCK / 32)
static constexpr float kSigma2 = 0.05f * 0.05f;

typedef float v2f __attribute__((ext_vector_type(2)));
typedef float v8f __attribute__((ext_vector_type(8)));
typedef __attribute__((address_space(3))) const char lds_cchar_t;

__device__ __forceinline__ void atomic_add_f32(float* p, float v) {
  __hip_atomic_fetch_add(p, v, __ATOMIC_RELAXED, __HIP_MEMORY_SCOPE_AGENT);
}

__device__ __forceinline__ unsigned lds_addr_of(const void* p) {
  return (unsigned)(uintptr_t)(lds_cchar_t*)p;
}

// CDNA5 async LDS staging (ASYNCcnt path, CDNA5 ISA)
__device__ __forceinline__ void async_copy_b32(void* lds_dst, const void* gsrc) {
  asm volatile("global_load_async_to_lds_b32 %0, %1, off"
               :: "v"(lds_addr_of(lds_dst)),
                  "v"((unsigned long long)(uintptr_t)gsrc)
               : "memory");
}
__device__ __forceinline__ void async_copy_b64(void* lds_dst, const void* gsrc) {
  asm volatile("global_load_async_to_lds_b64 %0, %1, off"
               :: "v"(lds_addr_of(lds_dst)),
                  "v"((unsigned long long)(uintptr_t)gsrc)
               : "memory");
}
__device__ __forceinline__ void wait_async0() {
  asm volatile("s_wait_asynccnt 0" ::: "memory");
}
__device__ __forceinline__ void lds_fence() {
  __builtin_amdgcn_wave_barrier();
  asm volatile("s_wait_dscnt 0" ::: "memory");
}

// ---------------- init: zero accumulators ----------------
__global__ void k_zero(float* __restrict__ p, int n) {
  int i = blockIdx.x * BLOCK + threadIdx.x;
  if (i < n) p[i] = 0.0f;
}

// ---- pass 1 (edges): neighbor counts + src/tgt sums per dst node ----
__global__ void k_edge_sums(const long long* __restrict__ ei,
                            const float* __restrict__ srcp,
                            const float* __restrict__ tgtp,
                            float* __restrict__ cnt,
                            float* __restrict__ ssum,
                            float* __restrict__ tsum,
                            int E) {
  __shared__ long long s_dst[BLOCK];
  __shared__ long long s_nbr[BLOCK];
  int tid = threadIdx.x;
  int e = blockIdx.x * BLOCK + tid;
  if (e < E) {
    // contiguous index streams staged via the async engine
    async_copy_b64(&s_dst[tid], ei + e);
    async_copy_b64(&s_nbr[tid], ei + (size_t)E + e);
    __builtin_prefetch(ei + e + 16 * BLOCK, 0, 0);
    __builtin_prefetch(ei + (size_t)E + e + 16 * BLOCK, 0, 0);
  }
  wait_async0();
  if (e >= E) return;
  int dst = (int)s_dst[tid];
  int nbr = (int)s_nbr[tid];
  float sx = srcp[3 * nbr + 0], sy = srcp[3 * nbr + 1], sz = srcp[3 * nbr + 2];
  float tx = tgtp[3 * nbr + 0], ty = tgtp[3 * nbr + 1], tz = tgtp[3 * nbr + 2];
  atomic_add_f32(&cnt[dst], 1.0f);
  atomic_add_f32(&ssum[3 * dst + 0], sx);
  atomic_add_f32(&ssum[3 * dst + 1], sy);
  atomic_add_f32(&ssum[3 * dst + 2], sz);
  atomic_add_f32(&tsum[3 * dst + 0], tx);
  atomic_add_f32(&tsum[3 * dst + 1], ty);
  atomic_add_f32(&tsum[3 * dst + 2], tz);
}

// ---- pass 2 (nodes): centers = sums / max(count, 1) ----
__global__ void k_centers(const float* __restrict__ cnt,
                          const float* __restrict__ ssum,
                          const float* __restrict__ tsum,
                          float* __restrict__ scent,
                          float* __restrict__ tcent,
                          int N) {
  int n = blockIdx.x * BLOCK + threadIdx.x;
  if (n >= N) return;
  float inv = 1.0f / fmaxf(cnt[n], 1.0f);
  scent[3 * n + 0] = ssum[3 * n + 0] * inv;
  scent[3 * n + 1] = ssum[3 * n + 1] * inv;
  scent[3 * n + 2] = ssum[3 * n + 2] * inv;
  tcent[3 * n + 0] = tsum[3 * n + 0] * inv;
  tcent[3 * n + 1] = tsum[3 * n + 1] * inv;
  tcent[3 * n + 2] = tsum[3 * n + 2] * inv;
}

// ---- pass 3 (edges): weighted outer-product scatter into per-node 3x3 cov ----
__global__ void k_edge_cov(const long long* __restrict__ ei,
                           const float* __restrict__ srcp,
                           const float* __restrict__ tgtp,
                           const float* __restrict__ ew,
                           const float* __restrict__ scent,
                           const float* __restrict__ tcent,
                           float* __restrict__ cov,
                           int E) {
  __shared__ long long s_dst[BLOCK];
  __shared__ long long s_nbr[BLOCK];
  __shared__ float s_w[BLOCK];
  int tid = threadIdx.x;
  int e = blockIdx.x * BLOCK + tid;
  if (e < E) {
    async_copy_b64(&s_dst[tid], ei + e);
    async_copy_b64(&s_nbr[tid], ei + (size_t)E + e);
    async_copy_b32(&s_w[tid], ew + e);
    __builtin_prefetch(ei + e + 16 * BLOCK, 0, 0);
    __builtin_prefetch(ei + (size_t)E + e + 16 * BLOCK, 0, 0);
    __builtin_prefetch(ew + e + 16 * BLOCK, 0, 0);
  }
  wait_async0();
  if (e >= E) return;

  float w = s_w[tid];
  int dst = (int)s_dst[tid];
  int nbr = (int)s_nbr[tid];

  float sdx = (srcp[3 * nbr + 0] - scent[3 * dst + 0]) * w;
  float sdy = (srcp[3 * nbr + 1] - scent[3 * dst + 1]) * w;
  float sdz = (srcp[3 * nbr + 2] - scent[3 * dst + 2]) * w;
  float tdx = tgtp[3 * nbr + 0] - tcent[3 * dst + 0];
  float tdy = tgtp[3 * nbr + 1] - tcent[3 * dst + 1];
  float tdz = tgtp[3 * nbr + 2] - tcent[3 * dst + 2];

  float* c = cov + (size_t)9 * dst;
  atomic_add_f32(c + 0, sdx * tdx);
  atomic_add_f32(c + 1, sdx * tdy);
  atomic_add_f32(c + 2, sdx * tdz);
  atomic_add_f32(c + 3, sdy * tdx);
  atomic_add_f32(c + 4, sdy * tdy);
  atomic_add_f32(c + 5, sdy * tdz);
  atomic_add_f32(c + 6, sdz * tdx);
  atomic_add_f32(c + 7, sdz * tdy);
  atomic_add_f32(c + 8, sdz * tdz);
}

// ---------------- 3x3 SVD helpers (per-node, scalar VALU) ----------------
__device__ __forceinline__ void jrot(float A[3][3], float V[3][3], int p, int q) {
  float apq = A[p][q];
  if (fabsf(apq) < 1e-30f) return;
  float theta = (A[q][q] - A[p][p]) / (2.0f * apq);
  float t = copysignf(1.0f, theta) / (fabsf(theta) + sqrtf(theta * theta + 1.0f));
  float c = rsqrtf(t * t + 1.0f);
  float s = t * c;
#pragma unroll
  for (int k = 0; k < 3; ++k) {
    float akp = A[k][p], akq = A[k][q];
    A[k][p] = c * akp - s * akq;
    A[k][q] = s * akp + c * akq;
  }
#pragma unroll
  for (int k = 0; k < 3; ++k) {
    float apk = A[p][k], aqk = A[q][k];
    A[p][k] = c * apk - s * aqk;
    A[q][k] = s * apk + c * aqk;
  }
#pragma unroll
  for (int k = 0; k < 3; ++k) {
    float vkp = V[k][p], vkq = V[k][q];
    V[k][p] = c * vkp - s * vkq;
    V[k][q] = s * vkp + c * vkq;
  }
}

__device__ __forceinline__ float dot3(const float* a, const float* b) {
  return a[0] * b[0] + a[1] * b[1] + a[2] * b[2];
}

// ---- pass 4 (nodes): SVD -> proper rotation R, translation t ----
// The per-node products R0 = U * V^T are evaluated 5-nodes-at-a-time with
// V_WMMA_F32_16X16X4_F32 (full f32): A[16x4] stacks five U blocks in rows,
// B[4x16] stacks five V^T blocks in columns over shared K=3; the diagonal
// 3x3 blocks of C are the five per-node products.
__global__ void k_svd(const float* __restrict__ cov,
                      const float* __restrict__ scent,
                      const float* __restrict__ tcent,
                      float* __restrict__ Rout,
                      float* __restrict__ tout,
                      int N) {
  __shared__ float lds_U[WAVES_PER_BLOCK][32][9];  // [k*3+i] = u_k[i]
  __shared__ float lds_V[WAVES_PER_BLOCK][32][9];  // [k*3+j] = v_k[j]
  __shared__ float lds_C[WAVES_PER_BLOCK][16][16];

  int tid = threadIdx.x;
  int wv = tid >> 5;
  int l = tid & 31;
  int n = blockIdx.x * BLOCK + tid;
  if (n >= N) n = N - 1;  // keep EXEC all-ones for WMMA (duplicates benign)

  float M[3][3];
#pragma unroll
  for (int i = 0; i < 3; ++i)
#pragma unroll
    for (int j = 0; j < 3; ++j) M[i][j] = cov[(size_t)9 * n + 3 * i + j];

  // S = M^T M (symmetric PSD)
  float A[3][3];
#pragma unroll
  for (int i = 0; i < 3; ++i)
#pragma unroll
    for (int j = 0; j < 3; ++j)
      A[i][j] = M[0][i] * M[0][j] + M[1][i] * M[1][j] + M[2][i] * M[2][j];

  float V[3][3] = {{1, 0, 0}, {0, 1, 0}, {0, 0, 1}};
#pragma unroll
  for (int sweep = 0; sweep < 6; ++sweep) {
    jrot(A, V, 0, 1);
    jrot(A, V, 0, 2);
    jrot(A, V, 1, 2);
  }

  // sort eigenpairs descending
  int i0 = 0, i1 = 1, i2 = 2, tswap;
  if (A[i0][i0] < A[i1][i1]) { tswap = i0; i0 = i1; i1 = tswap; }
  if (A[i0][i0] < A[i2][i2]) { tswap = i0; i0 = i2; i2 = tswap; }
  if (A[i1][i1] < A[i2][i2]) { tswap = i1; i1 = i2; i2 = tswap; }

  float vv[3][3];  // vv[k] = k-th right singular vector
#pragma unroll
  for (int r = 0; r < 3; ++r) {
    vv[0][r] = V[r][i0];
    vv[1][r] = V[r][i1];
    vv[2][r] = V[r][i2];
  }
  float s0 = sqrtf(fmaxf(A[i0][i0], 0.0f));
  float s2 = sqrtf(fmaxf(A[i2][i2], 0.0f));

  // u_k = M v_k, then orthonormalize (Gram-Schmidt keeps det sign)
  float u[3][3];
#pragma unroll
  for (int k = 0; k < 3; ++k)
#pragma unroll
    for (int i = 0; i < 3; ++i)
      u[k][i] = M[i][0] * vv[k][0] + M[i][1] * vv[k][1] + M[i][2] * vv[k][2];

  float n0 = sqrtf(dot3(u[0], u[0]));
  if (n0 > 1e-20f) {
    float r = 1.0f / n0;
    u[0][0] *= r; u[0][1] *= r; u[0][2] *= r;
  } else {
    u[0][0] = 1.0f; u[0][1] = 0.0f; u[0][2] = 0.0f;
  }

  float d01 = dot3(u[1], u[0]);
#pragma unroll
  for (int i = 0; i < 3; ++i) u[1][i] -= d01 * u[0][i];
  float n1 = sqrtf(dot3(u[1], u[1]));
  if (n1 > 1e-20f) {
    float r = 1.0f / n1;
    u[1][0] *= r; u[1][1] *= r; u[1][2] *= r;
  } else {
    float ax = fabsf(u[0][0]), ay = fabsf(u[0][1]), az = fabsf(u[0][2]);
    float e0 = 0.0f, e1 = 0.0f, e2 = 0.0f;
    if (ax <= ay && ax <= az) e0 = 1.0f;
    else if (ay <= az) e1 = 1.0f;
    else e2 = 1.0f;
    float cx = u[0][1] * e2 - u[0][2] * e1;
    float cy = u[0][2] * e0 - u[0][0] * e2;
    float cz = u[0][0] * e1 - u[0][1] * e0;
    float cn = rsqrtf(cx * cx + cy * cy + cz * cz + 1e-30f);
    u[1][0] = cx * cn; u[1][1] = cy * cn; u[1][2] = cz * cn;
  }

  float d02 = dot3(u[2], u[0]);
  float d12 = dot3(u[2], u[1]);
#pragma unroll
  for (int i = 0; i < 3; ++i) u[2][i] -= d02 * u[0][i] + d12 * u[1][i];
  float n2 = sqrtf(dot3(u[2], u[2]));
  bool full_rank = (s2 > 1e-6f * (s0 + 1e-30f)) && (n2 > 1e-20f);
  if (full_rank) {
    float r = 1.0f / n2;
    u[2][0] *= r; u[2][1] *= r; u[2][2] *= r;
  } else {
    u[2][0] = u[0][1] * u[1][2] - u[0][2] * u[1][1];
    u[2][1] = u[0][2] * u[1][0] - u[0][0] * u[1][2];
    u[2][2] = u[0][0] * u[1][1] - u[0][1] * u[1][0];
  }

  // ---- publish U/V, compute R0 = U V^T via WMMA, 5 nodes per issue ----
#pragma unroll
  for (int k = 0; k < 3; ++k)
#pragma unroll
    for (int i = 0; i < 3; ++i) {
      lds_U[wv][l][k * 3 + i] = u[k][i];
      lds_V[wv][l][k * 3 + i] = vv[k][i];
    }
  lds_fence();

  float R0[3][3];
  const int mrow = l & 15;           // A row / C row base, and B col / C col
  const int kb = (l < 16) ? 0 : 2;   // this lane's K pair
  const int mofs = (l < 16) ? 0 : 8; // C-layout row offset
  const int na_node = mrow / 3, na_r = mrow % 3;  // A: row -> (node, row-in-block)

  for (int b = 0; b < 7; ++b) {
    int anode = b * 5 + na_node;
    v2f av, bv;
    // A[mrow][kb], A[mrow][kb+1] = u_{anode}[k][row]
    av.x = (mrow < 15 && anode < 32) ? lds_U[wv][anode][kb * 3 + na_r] : 0.0f;
    av.y = (mrow < 15 && anode < 32 && (kb + 1) < 3)
               ? lds_U[wv][anode][(kb + 1) * 3 + na_r] : 0.0f;
    // B[kb][ncol], B[kb+1][ncol] = v_{bnode}[k][col]
    bv.x = (mrow < 15 && anode < 32) ? lds_V[wv][anode][kb * 3 + na_r] : 0.0f;
    bv.y = (mrow < 15 && anode < 32 && (kb + 1) < 3)
               ? lds_V[wv][anode][(kb + 1) * 3 + na_r] : 0.0f;
    v8f cacc = {};
    cacc = __builtin_amdgcn_wmma_f32_16x16x4_f32(
        false, av, false, bv, (short)0, cacc, false, false);
    // scatter D (16x16 f32 layout: VGPR r, lane l -> C[r + mofs][l & 15])
#pragma unroll
    for (int r = 0; r < 8; ++r) lds_C[wv][mofs + r][mrow] = cacc[r];
    lds_fence();
    int t = l - b * 5;
    if (t >= 0 && t < 5) {
#pragma unroll
      for (int i = 0; i < 3; ++i)
#pragma unroll
        for (int j = 0; j < 3; ++j) R0[i][j] = lds_C[wv][3 * t + i][3 * t + j];
    }
    lds_fence();
  }

  float det0 = R0[0][0] * (R0[1][1] * R0[2][2] - R0[1][2] * R0[2][1])
             - R0[0][1] * (R0[1][0] * R0[2][2] - R0[1][2] * R0[2][0])
             + R0[0][2] * (R0[1][0] * R0[2][1] - R0[1][1] * R0[2][0]);

  float R[3][3];
#pragma unroll
  for (int i = 0; i < 3; ++i)
#pragma unroll
    for (int j = 0; j < 3; ++j)
      R[i][j] = R0[i][j] + (det0 - 1.0f) * u[2][i] * vv[2][j];

  float scx = scent[3 * n + 0], scy = scent[3 * n + 1], scz = scent[3 * n + 2];
#pragma unroll
  for (int i = 0; i < 3; ++i) {
    Rout[(size_t)9 * n + 3 * i + 0] = R[i][0];
    Rout[(size_t)9 * n + 3 * i + 1] = R[i][1];
    Rout[(size_t)9 * n + 3 * i + 2] = R[i][2];
    tout[3 * n + i] =
        tcent[3 * n + i] - (R[i][0] * scx + R[i][1] * scy + R[i][2] * scz);
  }
}

// ---- pass 5 (edges): apply neighbor transform, robust reweight ----
__global__ void k_reweight(const long long* __restrict__ ei,
                           const float* __restrict__ srcp,
                           const float* __restrict__ tgtp,
                           const float* __restrict__ Rn,
                           const float* __restrict__ tn,
                           float* __restrict__ wout,
                           int E) {
  __shared__ long long s_nbr[BLOCK];
  int tid = threadIdx.x;
  int e = blockIdx.x * BLOCK + tid;
  if (e < E) {
    async_copy_b64(&s_nbr[tid], ei + (size_t)E + e);
    __builtin_prefetch(ei + (size_t)E + e + 16 * BLOCK, 0, 0);
  }
  wait_async0();
  if (e >= E) return;
  int nbr = (int)s_nbr[tid];
  const float* R = Rn + (size_t)9 * nbr;
  float sx = srcp[3 * nbr + 0], sy = srcp[3 * nbr + 1], sz = srcp[3 * nbr + 2];
  float px = R[0] * sx + R[1] * sy + R[2] * sz + tn[3 * nbr + 0] - tgtp[3 * nbr + 0];
  float py = R[3] * sx + R[4] * sy + R[5] * sz + tn[3 * nbr + 1] - tgtp[3 * nbr + 1];
  float pz = R[6] * sx + R[7] * sy + R[8] * sz + tn[3 * nbr + 2] - tgtp[3 * nbr + 2];
  float d2 = px * px + py * py + pz * pz;
  wout[e] = kSigma2 / (d2 + kSigma2);
}

extern "C" void kernel_launch(void* const* d_in, const int* in_sizes, int n_in,
                              void* d_out, int out_size, void* d_ws, size_t ws_size,
                              hipStream_t stream) {
  const float* srcp = (const float*)d_in[0];
  const float* tgtp = (const float*)d_in[1];
  const long long* ei = (const long long*)d_in[2];  // int64 [2,E]
  const float* ew = (const float*)d_in[3];          // [E,1]
  const int N = in_sizes[0] / 3;
  const int E = in_sizes[2] / 2;

  // workspace (floats): cnt N | ssum 3N | tsum 3N | cov 9N | scent 3N | tcent 3N
  float* ws = (float*)d_ws;
  float* cnt = ws;
  float* ssum = cnt + N;
  float* tsum = ssum + 3 * (size_t)N;
  float* cov = tsum + 3 * (size_t)N;
  float* scent = cov + 9 * (size_t)N;
  float* tcent = scent + 3 * (size_t)N;

  // output (floats): R 9N | trans 3N | new_edge_weights E
  float* Rout = (float*)d_out;
  float* tout = Rout + 9 * (size_t)N;
  float* wout = tout + 3 * (size_t)N;

  const int nAccum = 16 * N;  // cnt + ssum + tsum + cov (contiguous)
  const int gE = (E + BLOCK - 1) / BLOCK;
  const int gN = (N + BLOCK - 1) / BLOCK;

  k_zero<<<(nAccum + BLOCK - 1) / BLOCK, BLOCK, 0, stream>>>(ws, nAccum);
  k_edge_sums<<<gE, BLOCK, 0, stream>>>(ei, srcp, tgtp, cnt, ssum, tsum, E);
  k_centers<<<gN, BLOCK, 0, stream>>>(cnt, ssum, tsum, scent, tcent, N);
  k_edge_cov<<<gE, BLOCK, 0, stream>>>(ei, srcp, tgtp, ew, scent, tcent, cov, E);
  k_svd<<<gN, BLOCK, 0, stream>>>(cov, scent, tcent, Rout, tout, N);
  k_reweight<<<gE, BLOCK, 0, stream>>>(ei, srcp, tgtp, Rout, tout, wout, E);
}